// PhysicsLoss_65687229825962
// MI455X (gfx1250) — compile-verified
//
#include <hip/hip_runtime.h>
#include <hip/hip_bf16.h>

// ---------------------------------------------------------------------------
// Physics loss for damped harmonic oscillator rollouts (B=4096, T=2048).
//
// Strategy (MI455X / gfx1250):
//  * The per-batch scan is LINEAR: [q;pi]' = A [q;pi], A 2x2. One wave per
//    batch; lane L starts at state A^(64L) x0 (6 squarings -> A^64, then
//    branchless binary exponentiation for (A^64)^L) and rolls 64 steps.
//    Serial depth 2048 -> 64, waves 128 -> 4096.
//  * x[b,:,:] is staged into LDS with the Tensor Data Mover: 3D tile
//    (2 floats) x (32 lane-chunks, stride 512B) x (16 steps, stride 8B),
//    landing time-major in LDS => conflict-free ds_load_b64 per lane.
//    Double buffered on TENSORcnt.
//  * Deterministic reduction: wave shfl tree -> per-batch partials in d_ws
//    -> single-block finalize kernel (double precision accumulate).
// ---------------------------------------------------------------------------

#ifndef BATCHES
#define BATCHES 4096
#endif
#define T_STEPS 2048
#define ROW_FLOATS (T_STEPS * 2)   // 4096 floats per batch row
#define CHUNK 64                    // timesteps per lane
#define JSTEP 16                    // timesteps per TDM tile (4 tiles/lane)

typedef unsigned int u32x4 __attribute__((ext_vector_type(4)));
typedef int i32x4 __attribute__((ext_vector_type(4)));
typedef int i32x8 __attribute__((ext_vector_type(8)));

template <int N> struct IC { static constexpr int value = N; };

struct M2 { float a, b, c, d; };

__device__ __forceinline__ M2 mmul(const M2 X, const M2 Y) {
  M2 r;
  r.a = fmaf(X.a, Y.a, X.b * Y.c);
  r.b = fmaf(X.a, Y.b, X.b * Y.d);
  r.c = fmaf(X.c, Y.a, X.d * Y.c);
  r.d = fmaf(X.c, Y.b, X.d * Y.d);
  return r;
}

// Issue one TDM load: tile = 2 x 32 x 16 f32 elements.
//   dim0 (x): 2 contiguous floats (q,pi pair)
//   dim1 (y): 32 lane-chunks, stride 128 floats (= 64 timesteps)
//   dim2 (z): 16 timesteps,   stride 2 floats
// LDS receives x-fastest,then y,then z  ==> layout [j][lane][2] (time-major).
__device__ __forceinline__ void tdm_load_tile(const float* gsrc, unsigned lds_off) {
  unsigned long long ga = (unsigned long long)(size_t)gsrc;
  u32x4 g0;
  g0[0] = 1u;                                   // count = 1 (valid descriptor)
  g0[1] = lds_off;                              // LDS byte address
  g0[2] = (unsigned)ga;                         // global addr [31:0]
  g0[3] = (unsigned)((ga >> 32) & 0x01FFFFFFu)  // global addr [56:32]
          | (2u << 30);                         // type = 2 ("image")
  i32x8 g1;
  g1[0] = 0x00020000;       // workgroup_mask=0 (not in cluster), data_size=2 (4B)
  g1[1] = 2 << 16;          // tensor_dim0 = 2
  g1[2] = 32 << 16;         // tensor_dim1 = 32
  g1[3] = 2 << 16;          // tile_dim0 = 2
  g1[4] = 32 | (16 << 16);  // tile_dim1 = 32, tile_dim2 = 16
  g1[5] = 128;              // tensor_dim0_stride = 128 floats (64 timesteps)
  g1[6] = 2 << 16;          // tensor_dim1_stride = 2 floats (1 timestep)
  g1[7] = 0;
  i32x4 g2; g2[0] = 16; g2[1] = 0; g2[2] = 0; g2[3] = 0;  // tensor_dim2 = 16
  i32x4 g3; g3[0] = 0;  g3[1] = 0; g3[2] = 0; g3[3] = 0;
  i32x8 g4; g4[0] = 0; g4[1] = 0; g4[2] = 0; g4[3] = 0;   // unused operand group
  g4[4] = 0; g4[5] = 0; g4[6] = 0; g4[7] = 0;             // (zero-filled)
  __builtin_amdgcn_tensor_load_to_lds(g0, g1, g2, g3, g4, 0);
}

__global__ __launch_bounds__(32) void rollout_loss_kernel(
    const float* __restrict__ x, const float* __restrict__ y,
    float* __restrict__ partial) {
  __shared__ float tile[2][JSTEP * 32 * 2];   // 2 x 4KB double buffer
  const int b    = blockIdx.x;
  const int lane = threadIdx.x;
  const float* xrow = x + (size_t)b * ROW_FLOATS;

  const float kDT  = 0.01f;
  const float m    = y[3 * b + 0] + 0.1f;
  const float kk   = y[3 * b + 1];
  const float lam  = y[3 * b + 2];
  const float dtm  = kDT / m;

  // One-step update matrix A (exact linearization of the reference step).
  M2 A; A.a = 1.0f; A.b = dtm; A.c = -kDT * kk;
  A.d = 1.0f - kDT * lam - kDT * dtm * kk;

  // Bm = A^64 via 6 squarings (uniform across the wave).
  M2 Bm = A;
#pragma unroll
  for (int i = 0; i < 6; ++i) Bm = mmul(Bm, Bm);

  // Lane L: Mm = Bm^L via branchless binary exponentiation (bits 0..4 of L).
  M2 Mm; Mm.a = 1.f; Mm.b = 0.f; Mm.c = 0.f; Mm.d = 1.f;
  M2 P = Bm;
#pragma unroll
  for (int i = 0; i < 5; ++i) {
    const bool bit = (lane >> i) & 1;
    M2 Q;
    Q.a = bit ? P.a : 1.f; Q.b = bit ? P.b : 0.f;
    Q.c = bit ? P.c : 0.f; Q.d = bit ? P.d : 1.f;
    Mm = mmul(Mm, Q);
    if (i < 4) P = mmul(P, P);
  }

  const float q0 = xrow[0], pi0 = xrow[1];
  float q  = fmaf(Mm.b, pi0, Mm.a * q0);   // state at t = 64*lane
  float pi = fmaf(Mm.d, pi0, Mm.c * q0);

  // Prime the TDM pipeline: tile s covers t = 64*L + 16*s + j, origin 32*s floats.
  tdm_load_tile(xrow + 0,  (unsigned)(size_t)&tile[0][0]);
  tdm_load_tile(xrow + 32, (unsigned)(size_t)&tile[1][0]);

  float accq = 0.f, accpi = 0.f;
  const float wbase = (float)(T_STEPS - CHUNK * lane);   // w[t] = 2048 - t

  // Superstep S: wait for tile S, consume 16 timesteps, reissue the buffer.
  // S must be a compile-time constant (s_wait_tensorcnt takes an ICE).
  auto superstep = [&](auto sc) {
    constexpr int S = decltype(sc)::value;
    __builtin_amdgcn_s_wait_tensorcnt(S == 3 ? 0 : 1);
    // Launder the LDS pointer so reads can't be folded (TDM wrote the data).
    unsigned long long bp = (unsigned long long)(size_t)&tile[S & 1][0];
    asm volatile("" : "+v"(bp) :: "memory");
    const float2* bufp = (const float2*)(size_t)bp;
#pragma unroll
    for (int j = 0; j < JSTEP; ++j) {
      if (!(S == 0 && j == 0)) {                      // first compare is at t=64L, no step
        q  = fmaf(dtm, pi, q);                        // q' = q + dt*pi/m
        pi = fmaf(-kDT, fmaf(kk, q, lam * pi), pi);   // pi' = pi - dt*(k q' + lam pi)
      }
      const float2 v = bufp[j * 32 + lane];           // ds_load_b64, conflict-free
      const float dq = q - v.x;
      const float dp = pi - v.y;
      const float wt = wbase - (float)(JSTEP * S + j);
      accq  = fmaf(wt * dq, dq, accq);
      accpi = fmaf(wt * dp, dp, accpi);
    }
    if constexpr (S < 2) {
      // Buffer S&1 is about to be overwritten: ensure our LDS reads retired.
      asm volatile("s_wait_dscnt 0" ::: "memory");
      tdm_load_tile(xrow + 32 * (S + 2), (unsigned)(size_t)&tile[S & 1][0]);
    }
  };
  superstep(IC<0>{});
  superstep(IC<1>{});
  superstep(IC<2>{});
  superstep(IC<3>{});

  // In-wave reduction (wave32).
#pragma unroll
  for (int off = 16; off > 0; off >>= 1) {
    accq  += __shfl_xor(accq,  off, 32);
    accpi += __shfl_xor(accpi, off, 32);
  }
  if (lane == 0) {
    partial[b]           = accq;
    partial[BATCHES + b] = accpi;
  }
}

__global__ __launch_bounds__(256) void finalize_kernel(
    const float* __restrict__ partial, const float* __restrict__ y,
    float* __restrict__ out) {
  __shared__ double sq[256], sp[256], pn[256];
  const int t = threadIdx.x;
  double aq = 0.0, ap = 0.0, ape = 0.0;
  for (int i = t; i < BATCHES; i += 256) {
    aq += (double)partial[i];
    ap += (double)partial[BATCHES + i];
  }
  for (int i = t; i < BATCHES * 3; i += 256) {
    const float v = y[i];
    if (v < 0.0f) ape += exp(-10.0 * (double)v);
  }
  sq[t] = aq; sp[t] = ap; pn[t] = ape;
  __syncthreads();
  for (int off = 128; off > 0; off >>= 1) {
    if (t < off) { sq[t] += sq[t + off]; sp[t] += sp[t + off]; pn[t] += pn[t + off]; }
    __syncthreads();
  }
  if (t == 0) {
    double rq = sqrt(sq[0] / (double)BATCHES); rq = fmin(rq, 1.0e15);
    double rp = sqrt(sp[0] / (double)BATCHES); rp = fmin(rp, 1.0e15);
    double pen = fmin(pn[0] / (double)(BATCHES * 3), 1000.0);
    out[0] = (float)(0.5 * (rq + rp) + pen);
  }
}

extern "C" void kernel_launch(void* const* d_in, const int* in_sizes, int n_in,
                              void* d_out, int out_size, void* d_ws, size_t ws_size,
                              hipStream_t stream) {
  (void)in_sizes; (void)n_in; (void)out_size; (void)ws_size;
  const float* x = (const float*)d_in[0];   // [4096, 2048, 2] f32
  const float* y = (const float*)d_in[1];   // [4096, 3] f32
  float* out = (float*)d_out;               // scalar f32
  float* partial = (float*)d_ws;            // 2*4096 f32 partials (32 KB)

  rollout_loss_kernel<<<BATCHES, 32, 0, stream>>>(x, y, partial);
  finalize_kernel<<<1, 256, 0, stream>>>(partial, y, out);
}